// Net_56599079026986
// MI455X (gfx1250) — compile-verified
//
#include <hip/hip_runtime.h>
#include <hip/hip_bf16.h>

// ---------------------------------------------------------------------------
// APPNP for MI455X (gfx1250): FP32 WMMA GEMMs (A-reuse register blocking)
// + L2-resident edge scatter with FP32 global atomics
// ---------------------------------------------------------------------------

typedef float v2f __attribute__((ext_vector_type(2)));
typedef float v8f __attribute__((ext_vector_type(8)));

#define ALPHA 0.1f

// ---------------------------------------------------------------------------
// GEMM: Y[N x Ncols] = act(X[N x K] * W[K x Ncols] + bias)
// One wave computes a 16 x (16*CT) strip: the A fragment is loaded ONCE per
// K-step and fed to CT back-to-back V_WMMA_F32_16X16X4_F32 ops (A-reuse).
// A layout (ISA 7.12.2, 32-bit A 16x4): lane 0-15 -> M=lane, K={0,1};
//                                       lane 16-31 -> M=lane-16, K={2,3}.
// B layout (symmetric): N=lane&15, half-wave selects K pair.
// C/D layout: VGPR i -> row i (lanes 0-15) / row i+8 (lanes 16-31), col=lane&15.
// ---------------------------------------------------------------------------
template <bool RELU, bool DUAL, int CT>
__global__ __launch_bounds__(64) void wmma_gemm_kernel(
    const float* __restrict__ X, const float* __restrict__ W,
    const float* __restrict__ bias, float* __restrict__ Y,
    float* __restrict__ Y2, int K, int Ncols)
{
    const int lane     = threadIdx.x;                                   // 0..31
    const int rowBase  = blockIdx.x * 16;
    const int colGroup = blockIdx.y * blockDim.y + threadIdx.y;
    const int colBase  = colGroup * 16 * CT;

    const int m  = lane & 15;        // row (A) / col (B)
    const int kh = (lane >> 4) * 2;  // K sub-pair: 0 or 2
    const int hi = lane >> 4;

    const float* __restrict__ arow = X + (size_t)(rowBase + m) * K + kh;
    const float* __restrict__ bcol = W + (size_t)kh * Ncols + colBase + m;

    v8f acc[CT];
#pragma unroll
    for (int t = 0; t < CT; ++t) acc[t] = (v8f){};

#pragma unroll 4
    for (int k = 0; k < K; k += 4) {
        v2f a;
        a.x = arow[k];
        a.y = arow[k + 1];
        v2f b[CT];
#pragma unroll
        for (int t = 0; t < CT; ++t) {
            b[t].x = bcol[(size_t)k * Ncols + t * 16];
            b[t].y = bcol[(size_t)k * Ncols + Ncols + t * 16];
        }
#pragma unroll
        for (int t = 0; t < CT; ++t) {
            acc[t] = __builtin_amdgcn_wmma_f32_16x16x4_f32(
                /*neg_a=*/false, a, /*neg_b=*/false, b[t],
                /*c_mod=*/(short)0, acc[t], /*reuse_a=*/false, /*reuse_b=*/false);
        }
    }

#pragma unroll
    for (int t = 0; t < CT; ++t) {
        const int cb = colBase + t * 16;
        const float bv = bias[cb + m];
#pragma unroll
        for (int i = 0; i < 8; ++i) {
            const int    r = rowBase + i + 8 * hi;
            const size_t o = (size_t)r * Ncols + cb + m;
            float v = acc[t][i] + bv;
            if (RELU) v = fmaxf(v, 0.0f);
            Y[o] = v;
            if (DUAL) Y2[o] = v;
        }
    }
}

// ---------------------------------------------------------------------------
// Degree / normalization kernels
// ---------------------------------------------------------------------------
__global__ __launch_bounds__(256) void deg_init_kernel(float* __restrict__ deg, int N)
{
    int i = blockIdx.x * blockDim.x + threadIdx.x;
    if (i < N) deg[i] = 1.0f;   // self-loop contribution
}

__global__ __launch_bounds__(256) void deg_edge_kernel(const int* __restrict__ dst,
                                                       float* __restrict__ deg, int E)
{
    int e = blockIdx.x * blockDim.x + threadIdx.x;
    if (e < E) unsafeAtomicAdd(&deg[dst[e]], 1.0f);
}

__global__ __launch_bounds__(256) void dinv_kernel(float* __restrict__ deg, int N)
{
    int i = blockIdx.x * blockDim.x + threadIdx.x;
    if (i < N) {
        float d = deg[i];
        deg[i] = (d > 0.0f) ? rsqrtf(d) : 0.0f;   // in-place: deg -> dinv
    }
}

__global__ __launch_bounds__(256) void norm_kernel(const int* __restrict__ src,
                                                   const int* __restrict__ dst,
                                                   const float* __restrict__ dinv,
                                                   float* __restrict__ nrm, int E)
{
    int e = blockIdx.x * blockDim.x + threadIdx.x;
    if (e < E) nrm[e] = dinv[src[e]] * dinv[dst[e]];
}

// ---------------------------------------------------------------------------
// APPNP iteration kernels (C = 64 features per node)
// ---------------------------------------------------------------------------
// agg[i,:] = dinv[i]^2 * x[i,:]   (self-loop term, also serves as zero-init)
__global__ __launch_bounds__(256) void selfloop_init_kernel(
    const float* __restrict__ dinv, const float* __restrict__ x,
    float* __restrict__ agg, int total /* N*64 */)
{
    int idx = blockIdx.x * blockDim.x + threadIdx.x;
    if (idx < total) {
        float di = dinv[idx >> 6];
        agg[idx] = di * di * x[idx];
    }
}

// agg[dst,:] += norm[e] * x[src,:] — 16 threads/edge, float4 per thread.
__global__ __launch_bounds__(256) void scatter_kernel(
    const int* __restrict__ src, const int* __restrict__ dst,
    const float* __restrict__ nrm, const float* __restrict__ x,
    float* __restrict__ agg, int E)
{
    unsigned gid = blockIdx.x * blockDim.x + threadIdx.x;
    int e = (int)(gid >> 4);
    int c = (int)(gid & 15) * 4;
    if (e >= E) return;
    const int   s = src[e];
    const int   d = dst[e];
    const float w = nrm[e];
    const float4 xv = *(const float4*)(x + (size_t)s * 64 + c);
    float* a = agg + (size_t)d * 64 + c;
    unsafeAtomicAdd(a + 0, w * xv.x);
    unsafeAtomicAdd(a + 1, w * xv.y);
    unsafeAtomicAdd(a + 2, w * xv.z);
    unsafeAtomicAdd(a + 3, w * xv.w);
}

// x = (1-alpha)*agg + alpha*h0
__global__ __launch_bounds__(256) void appnp_update_kernel(
    float* __restrict__ x, const float* __restrict__ agg,
    const float* __restrict__ h0, int total)
{
    int idx = blockIdx.x * blockDim.x + threadIdx.x;
    if (idx < total)
        x[idx] = (1.0f - ALPHA) * agg[idx] + ALPHA * h0[idx];
}

// ---------------------------------------------------------------------------
// Row-wise log-softmax over C=64: one wave32 per row, 2 cols per lane.
// ---------------------------------------------------------------------------
__global__ __launch_bounds__(256) void logsoftmax_kernel(
    const float* __restrict__ X, float* __restrict__ out, int N)
{
    const int lane = threadIdx.x & 31;
    const int wave = threadIdx.x >> 5;
    const int row  = blockIdx.x * 8 + wave;
    if (row >= N) return;

    const float* x = X + (size_t)row * 64;
    float v0 = x[lane];
    float v1 = x[lane + 32];

    float mx = fmaxf(v0, v1);
#pragma unroll
    for (int off = 16; off > 0; off >>= 1)
        mx = fmaxf(mx, __shfl_xor(mx, off, 32));

    float s = __expf(v0 - mx) + __expf(v1 - mx);
#pragma unroll
    for (int off = 16; off > 0; off >>= 1)
        s += __shfl_xor(s, off, 32);

    const float lse = mx + __logf(s);
    float* o = out + (size_t)row * 64;
    o[lane]      = v0 - lse;
    o[lane + 32] = v1 - lse;
}

// ---------------------------------------------------------------------------
// Host side
// ---------------------------------------------------------------------------
extern "C" void kernel_launch(void* const* d_in, const int* in_sizes, int n_in,
                              void* d_out, int out_size, void* d_ws, size_t ws_size,
                              hipStream_t stream)
{
    constexpr int N = 100000, E = 3200000, F_IN = 256, H = 128, C = 64, K = 10;

    const float* X   = (const float*)d_in[0];  // [N, F_IN]
    const int*   ei  = (const int*)  d_in[1];  // [2, E] row-major: src then dst
    const float* W0  = (const float*)d_in[2];  // [F_IN, H]
    const float* b0  = (const float*)d_in[3];  // [H]
    const float* W1  = (const float*)d_in[4];  // [H, C]
    const float* b1  = (const float*)d_in[5];  // [C]
    float*       out = (float*)d_out;          // [N, C]

    const int* src = ei;
    const int* dst = ei + E;

    // Workspace layout (all float32, 16B-aligned blocks)
    char* ws = (char*)d_ws;
    float* h    = (float*)ws; ws += sizeof(float) * (size_t)N * H;   // 51.2 MB
    float* x0   = (float*)ws; ws += sizeof(float) * (size_t)N * C;   // 25.6 MB
    float* x    = (float*)ws; ws += sizeof(float) * (size_t)N * C;   // 25.6 MB
    float* agg  = (float*)ws; ws += sizeof(float) * (size_t)N * C;   // 25.6 MB
    float* dinv = (float*)ws; ws += sizeof(float) * (size_t)N;       //  0.4 MB
    float* nrm  = (float*)ws; ws += sizeof(float) * (size_t)E;       // 12.8 MB

    // 1) h = relu(X @ W0 + b0): 8 col tiles = 2 waves x CT=4, block(32,2)
    wmma_gemm_kernel<true, false, 4><<<dim3(N / 16, 1), dim3(32, 2), 0, stream>>>(
        X, W0, b0, h, nullptr, F_IN, H);

    // 2) x0 = h @ W1 + b1 ; x = x0 (dual write): 4 col tiles = 1 wave x CT=4
    wmma_gemm_kernel<false, true, 4><<<dim3(N / 16, 1), dim3(32, 1), 0, stream>>>(
        h, W1, b1, x0, x, H, C);

    // 3) gcn_norm: degree (self-loops + edges), dinv = rsqrt(deg), edge norms
    deg_init_kernel<<<(N + 255) / 256, 256, 0, stream>>>(dinv, N);
    deg_edge_kernel<<<(E + 255) / 256, 256, 0, stream>>>(dst, dinv, E);
    dinv_kernel<<<(N + 255) / 256, 256, 0, stream>>>(dinv, N);
    norm_kernel<<<(E + 255) / 256, 256, 0, stream>>>(src, dst, dinv, nrm, E);

    // 4) APPNP power iteration
    const int totalNC  = N * C;
    const int nBlksNC  = (totalNC + 255) / 256;
    const unsigned scatterThreads = (unsigned)E * 16u;
    const int nBlksSc  = (int)((scatterThreads + 255u) / 256u);
    for (int it = 0; it < K; ++it) {
        selfloop_init_kernel<<<nBlksNC, 256, 0, stream>>>(dinv, x, agg, totalNC);
        scatter_kernel<<<nBlksSc, 256, 0, stream>>>(src, dst, nrm, x, agg, E);
        appnp_update_kernel<<<nBlksNC, 256, 0, stream>>>(x, agg, x0, totalNC);
    }

    // 5) out = log_softmax(x, axis=1)
    logsoftmax_kernel<<<(N + 7) / 8, 256, 0, stream>>>(x, out, N);
}